// InitGCN_10531259810642
// MI455X (gfx1250) — compile-verified
//
#include <hip/hip_runtime.h>
#include <hip/hip_bf16.h>

typedef __attribute__((ext_vector_type(2))) float v2f;
typedef __attribute__((ext_vector_type(8))) float v8f;

#define HIDC 128

__device__ __forceinline__ void atomAddF32(float* p, float v) {
  // hardware global_atomic_add_f32 (L2-resident accumulators on MI455X)
  unsafeAtomicAdd(p, v);
}

// ---------------------------------------------------------------------------
// Dense GEMM via V_WMMA_F32_16X16X4_F32:  D[m][n] = sum_k A[m][k] * Bw[n][k]
// A: [nRows, K] row-major, Bw: [OUTF, K] row-major (torch Linear weight),
// optional bias[OUTF] and ReLU.
// One wave per 32-row tile (two 16-row WMMA tiles) so every B fragment is
// reused by two WMMAs -> halves B-side VMEM issue per FLOP. NT column tiles.
// ---------------------------------------------------------------------------
template <int K, int NT, int OUTF, bool RELU, bool BIAS>
__global__ __launch_bounds__(128) void gcn_wmma_gemm(
    const float* __restrict__ A, const float* __restrict__ Bw,
    const float* __restrict__ bias, float* __restrict__ D, int nRows) {
  const int wavesPerBlk = blockDim.x >> 5;
  const int tile = blockIdx.x * wavesPerBlk + (threadIdx.x >> 5);  // 32-row tile
  const int nTiles = (nRows + 31) >> 5;
  if (tile >= nTiles) return;  // wave-uniform exit

  const int lane = threadIdx.x & 31;
  const int half = lane >> 4;   // 0: K=k0,k0+1   1: K=k0+2,k0+3
  const int r    = lane & 15;   // row (A) / col (B) within the 16-wide tile

  int row0 = tile * 32 + r;        // first 16-row tile
  int row1 = tile * 32 + 16 + r;   // second 16-row tile
  if (row0 >= nRows) row0 = nRows - 1;  // defensive clamp (N % 32 == 0 here)
  if (row1 >= nRows) row1 = nRows - 1;
  const float* __restrict__ arow0 = A + (size_t)row0 * K + (half << 1);
  const float* __restrict__ arow1 = A + (size_t)row1 * K + (half << 1);

  v8f acc0[NT] = {};
  v8f acc1[NT] = {};

#pragma unroll 2
  for (int k0 = 0; k0 < K; k0 += 4) {
    v2f a0 = *(const v2f*)(arow0 + k0);
    v2f a1 = *(const v2f*)(arow1 + k0);
#pragma unroll
    for (int t = 0; t < NT; ++t) {
      const int n = t * 16 + r;
      v2f b = {0.0f, 0.0f};
      if (n < OUTF)
        b = *(const v2f*)(Bw + (size_t)n * K + k0 + (half << 1));
      // (neg_a, A, neg_b, B, c_mod, C, reuse_a, reuse_b)
      acc0[t] = __builtin_amdgcn_wmma_f32_16x16x4_f32(
          false, a0, false, b, (short)0, acc0[t], false, false);
      acc1[t] = __builtin_amdgcn_wmma_f32_16x16x4_f32(
          false, a1, false, b, (short)0, acc1[t], false, false);
    }
  }

  // D layout: VGPR m -> row m (lanes 0-15) / row m+8 (lanes 16-31), col = lane%16
  const int rowBase0 = tile * 32 + (half << 3);
  const int rowBase1 = rowBase0 + 16;
#pragma unroll
  for (int t = 0; t < NT; ++t) {
    const int n = t * 16 + r;
    if (n >= OUTF) continue;
    const float bv = BIAS ? bias[n] : 0.0f;
#pragma unroll
    for (int m = 0; m < 8; ++m) {
      const int ra = rowBase0 + m;
      const int rb = rowBase1 + m;
      if (ra < nRows) {
        float v = acc0[t][m] + bv;
        if (RELU) v = fmaxf(v, 0.0f);
        D[(size_t)ra * OUTF + n] = v;
      }
      if (rb < nRows) {
        float v = acc1[t][m] + bv;
        if (RELU) v = fmaxf(v, 0.0f);
        D[(size_t)rb * OUTF + n] = v;
      }
    }
  }
}

// ---------------------------------------------------------------------------
// Degree / normalization kernels
// ---------------------------------------------------------------------------
__global__ void deg_init_k(float* __restrict__ deg, int n) {
  int i = blockIdx.x * blockDim.x + threadIdx.x;
  if (i < n) deg[i] = 1.0f;  // self-loop contribution
}

__global__ void deg_accum_k(const long long* __restrict__ dst,
                            float* __restrict__ deg, int e) {
  int i = blockIdx.x * blockDim.x + threadIdx.x;
  if (i < e) atomAddF32(deg + dst[i], 1.0f);
}

__global__ void rsqrt_k(float* __restrict__ deg, int n) {
  int i = blockIdx.x * blockDim.x + threadIdx.x;
  if (i < n) deg[i] = rsqrtf(deg[i]);
}

// ---------------------------------------------------------------------------
// agg[i][c] = t[i][c] * dinv[i]^2   (self-loop term, also zero-initializes agg)
// ---------------------------------------------------------------------------
__global__ void self_init_k(const float* __restrict__ t,
                            const float* __restrict__ dinv,
                            float* __restrict__ agg, int nh) {
  int i = blockIdx.x * blockDim.x + threadIdx.x;
  if (i < nh) {
    const int node = i >> 7;  // /HIDC
    const float di = dinv[node];
    agg[i] = t[i] * di * di;
  }
}

// ---------------------------------------------------------------------------
// One wave per edge: agg[dst] += t[src] * dinv[src]*dinv[dst]
// lane handles 4 channels (float4). Both t and agg are L2-resident (51 MB).
// ---------------------------------------------------------------------------
__global__ __launch_bounds__(256) void edge_scatter_k(
    const float* __restrict__ t, const long long* __restrict__ srcI,
    const long long* __restrict__ dstI, const float* __restrict__ dinv,
    float* __restrict__ agg, int e) {
  const int w = blockIdx.x * (blockDim.x >> 5) + (threadIdx.x >> 5);
  if (w >= e) return;
  const int lane = threadIdx.x & 31;
  const long long s = srcI[w];
  const long long d = dstI[w];
  const float nm = dinv[s] * dinv[d];
  const float4 m = ((const float4*)(t + (size_t)s * HIDC))[lane];
  float* o = agg + (size_t)d * HIDC + lane * 4;
  atomAddF32(o + 0, m.x * nm);
  atomAddF32(o + 1, m.y * nm);
  atomAddF32(o + 2, m.z * nm);
  atomAddF32(o + 3, m.w * nm);
}

// ---------------------------------------------------------------------------
// h[i][c] = relu(agg[i][c] + b[c])   in place
// ---------------------------------------------------------------------------
__global__ void bias_relu_k(float* __restrict__ agg,
                            const float* __restrict__ b, int nh) {
  int i = blockIdx.x * blockDim.x + threadIdx.x;
  if (i < nh) agg[i] = fmaxf(agg[i] + b[i & (HIDC - 1)], 0.0f);
}

// ---------------------------------------------------------------------------
extern "C" void kernel_launch(void* const* d_in, const int* in_sizes, int n_in,
                              void* d_out, int out_size, void* d_ws,
                              size_t ws_size, hipStream_t stream) {
  const float*     x   = (const float*)d_in[0];
  const long long* ei  = (const long long*)d_in[1];
  const float*     W0  = (const float*)d_in[2];
  const float*     b0  = (const float*)d_in[3];
  const float*     Wc  = (const float*)d_in[4];
  const float*     bc  = (const float*)d_in[5];
  const float*     Wl  = (const float*)d_in[6];
  const float*     bl  = (const float*)d_in[7];
  float* out = (float*)d_out;

  const int INC = 512;
  const int N = in_sizes[0] / INC;
  const int E = in_sizes[1] / 2;

  // workspace: dinv | bufA | bufB   (buffers ping-pong; both fit in L2)
  float* dinv = (float*)d_ws;
  const size_t nAlign = ((size_t)N + 63) & ~(size_t)63;
  float* bufA = dinv + nAlign;
  float* bufB = bufA + (size_t)N * HIDC;

  const long long* srcI = ei;      // edge_index[0]
  const long long* dstI = ei + E;  // edge_index[1]

  // ---- symmetric GCN normalization -------------------------------------
  deg_init_k<<<(N + 255) / 256, 256, 0, stream>>>(dinv, N);
  deg_accum_k<<<(E + 255) / 256, 256, 0, stream>>>(dstI, dinv, E);
  rsqrt_k<<<(N + 255) / 256, 256, 0, stream>>>(dinv, N);

  const int tiles32 = (N + 31) / 32;           // 3125 for N=100000
  dim3 gemmGrid((tiles32 + 3) / 4);            // 4 waves / block

  // ---- input layer: h = relu(x @ W0^T + b0) ----------------------------
  gcn_wmma_gemm<512, 8, 128, true, true>
      <<<gemmGrid, 128, 0, stream>>>(x, W0, b0, bufA, N);

  // ---- 3 GCN conv layers ----------------------------------------------
  const int NH = N * HIDC;
  for (int l = 0; l < 3; ++l) {
    // t = h @ Wc[l]^T
    gcn_wmma_gemm<128, 8, 128, false, false><<<gemmGrid, 128, 0, stream>>>(
        bufA, Wc + (size_t)l * HIDC * HIDC, nullptr, bufB, N);
    // agg = self-loop term (overwrites now-dead h in bufA)
    self_init_k<<<(NH + 255) / 256, 256, 0, stream>>>(bufB, dinv, bufA, NH);
    // agg[dst] += t[src] * norm  (L2-resident atomics)
    edge_scatter_k<<<(E + 7) / 8, 256, 0, stream>>>(bufB, srcI, dstI, dinv,
                                                    bufA, E);
    // h = relu(agg + bc[l])
    bias_relu_k<<<(NH + 255) / 256, 256, 0, stream>>>(
        bufA, bc + (size_t)l * HIDC, NH);
  }

  // ---- output layer: logits = h @ Wl^T + bl ----------------------------
  gcn_wmma_gemm<128, 3, 40, false, true>
      <<<gemmGrid, 128, 0, stream>>>(bufA, Wl, bl, out, N);
}